// TNAttention_19559281066176
// MI455X (gfx1250) — compile-verified
//
#include <hip/hip_runtime.h>

// CDNA5 / gfx1250, wave32. Streaming-bound kernel: 268 MB of W_edges at
// 23.3 TB/s dominates (~11.5us floor); WMMA f32 16x16x4 does the per-edge
// 64x64 matvecs off the VALU with persistent register accumulators.
// W loads are non-temporal (one-pass stream > 192MB L2, zero reuse).

typedef float v2f __attribute__((ext_vector_type(2)));
typedef float v8f __attribute__((ext_vector_type(8)));

#define BOND 64
#define NEMB 1024
#define EDGE_BLOCKS 512
#define EDGE_THREADS 256
#define WAVES_PER_BLOCK (EDGE_THREADS / 32)
#define TOTAL_WAVES (EDGE_BLOCKS * WAVES_PER_BLOCK)

// ---------------- phase 0: acc = h = W_in @ x ----------------
__global__ void tn_h_init(const float* __restrict__ x,
                          const float* __restrict__ W_in,
                          float* __restrict__ acc) {
    int d = threadIdx.x;
    if (d < BOND) {
        const float* row = W_in + d * NEMB;
        float s = 0.0f;
        for (int i = 0; i < NEMB; ++i) s = fmaf(row[i], x[i], s);
        acc[d] = s;            // fresh init each launch (graph-replay safe)
    }
}

// ---------------- phase 1: acc += sum_e relu(g_e) * W_e @ h_src[e] ----------------
__launch_bounds__(EDGE_THREADS)
__global__ void tn_edges(const float* __restrict__ hidden_cache,
                         const int*   __restrict__ src,
                         const float* __restrict__ gates,
                         const float* __restrict__ W_edges,
                         float* __restrict__ acc,
                         int E) {
    __shared__ __align__(16) float lds_h[WAVES_PER_BLOCK][BOND]; // per-wave g*h
    __shared__ __align__(16) float lds_acc[BOND];

    const int lane = threadIdx.x & 31;
    const int wib  = threadIdx.x >> 5;
    const int wave = (blockIdx.x * EDGE_THREADS + (int)threadIdx.x) >> 5;

    if (threadIdx.x < BOND) lds_acc[threadIdx.x] = 0.0f;
    __syncthreads();

    // Persistent C accumulators: 4 row-tiles of 16 output rows each.
    v8f c[4] = {};

    const int koff = (lane >> 4) * 2;   // 0 for lanes 0-15, 2 for lanes 16-31
    const int mrow = lane & 15;         // M index within a 16-row tile

    for (int e = wave; e < E; e += TOTAL_WAVES) {
        float g = gates[e];             // wave-uniform -> EXEC stays all-ones
        if (g > 0.0f) {
            // gated source vector into this wave's private LDS slice
            const float* hsrc = hidden_cache + (long)src[e] * BOND;
            v2f hv = *(const v2f*)(hsrc + 2 * lane);
            v2f hg; hg.x = g * hv.x; hg.y = g * hv.y;
            *(v2f*)(&lds_h[wib][2 * lane]) = hg;        // ds_store_b64

            // Build all 16 B fragments (K-slices of 4) once; reused by 4 row tiles.
            // B layout (4x16): VGPR0 = rows K0/K2 (lane halves), VGPR1 = K1/K3;
            // broadcasting the same value across N makes every D column the matvec.
            v2f b[16];
            const float* hp = &lds_h[wib][koff];
            #pragma unroll
            for (int s = 0; s < 16; ++s)
                b[s] = *(const v2f*)(hp + 4 * s);        // ds_load_b64

            const float* W = W_edges + (long)e * (BOND * BOND);

            #pragma unroll
            for (int r = 0; r < 4; ++r) {
                // A layout (16x4): lane m holds {W[row,K0],W[row,K1]},
                // lane m+16 holds {W[row,K2],W[row,K3]} -> 2 consecutive floats.
                const float* Wr = W + (16 * r + mrow) * BOND + koff;
                #pragma unroll
                for (int s = 0; s < 16; ++s) {
                    // one-pass 268MB stream: non-temporal, don't thrash L2
                    v2f a = __builtin_nontemporal_load((const v2f*)(Wr + 4 * s));
                    c[r] = __builtin_amdgcn_wmma_f32_16x16x4_f32(
                        /*neg_a=*/false, a, /*neg_b=*/false, b[s],
                        /*c_mod=*/(short)0, c[r],
                        /*reuse_a=*/false, /*reuse_b=*/false);
                }
            }
        }
    }

    // C layout: VGPR j -> M=j (lanes 0-15), M=j+8 (lanes 16-31); all N equal.
    if (mrow == 0) {                    // lanes 0 and 16 carry the results
        int off = (lane >> 4) * 8;
        #pragma unroll
        for (int r = 0; r < 4; ++r)
            #pragma unroll
            for (int j = 0; j < 8; ++j)
                atomicAdd(&lds_acc[16 * r + off + j], c[r][j]);   // ds_add_f32
    }
    __syncthreads();
    if (threadIdx.x < BOND)
        atomicAdd(&acc[threadIdx.x], lds_acc[threadIdx.x]);       // global_atomic_add_f32
}

// ---------------- phase 2: out = W_out @ acc ----------------
__global__ void tn_out(const float* __restrict__ W_out,
                       const float* __restrict__ acc,
                       float* __restrict__ out) {
    int i = blockIdx.x * blockDim.x + threadIdx.x;
    if (i < NEMB) {
        const float* row = W_out + i * BOND;
        float s = 0.0f;
        #pragma unroll
        for (int d = 0; d < BOND; ++d) s = fmaf(row[d], acc[d], s);
        out[i] = s;
    }
}

extern "C" void kernel_launch(void* const* d_in, const int* in_sizes, int n_in,
                              void* d_out, int out_size, void* d_ws, size_t ws_size,
                              hipStream_t stream) {
    const float* x            = (const float*)d_in[0];
    const float* hidden_cache = (const float*)d_in[1];
    const int*   src          = (const int*)  d_in[2];
    const float* gates        = (const float*)d_in[3];
    const float* W_edges      = (const float*)d_in[4];
    const float* W_in         = (const float*)d_in[5];
    const float* W_out        = (const float*)d_in[6];
    float*       out          = (float*)d_out;
    float*       acc          = (float*)d_ws;          // 64 floats of scratch
    const int E = in_sizes[2];

    tn_h_init<<<1, 64, 0, stream>>>(x, W_in, acc);
    tn_edges<<<EDGE_BLOCKS, EDGE_THREADS, 0, stream>>>(hidden_cache, src, gates,
                                                       W_edges, acc, E);
    tn_out<<<(NEMB + 255) / 256, 256, 0, stream>>>(W_out, acc, out);
}